// WalkEncoder_85109071937616
// MI455X (gfx1250) — compile-verified
//
#include <hip/hip_runtime.h>
#include <hip/hip_bf16.h>

// ---------------- problem constants ----------------
#define NNODE 50000
#define NEDGE 100000
#define NWALK 20000
#define LWALK 50
#define HDIM  128
#define PEDIM 15
#define KCONV 9
#define EPSV  1e-5f

// padded tile geometry
#define ROWS  64      // walk rows padded to multiple of 16
#define SROW  132     // LDS row stride (floats); 132*4B = 33*16B -> float4-aligned rows
#define SPROW 20      // LDS row stride for 16-wide PE tile

typedef __attribute__((ext_vector_type(2))) float v2f;
typedef __attribute__((ext_vector_type(8))) float v8f;

__device__ __forceinline__ v8f wmma_f32x4(v2f a, v2f b, v8f c) {
  // D = A(16x4) x B(4x16) + C(16x16), all f32 -> v_wmma_f32_16x16x4_f32
  return __builtin_amdgcn_wmma_f32_16x16x4_f32(false, a, false, b, (short)0, c,
                                               false, false);
}

__device__ __forceinline__ void atomAddF(float* p, float v) {
  unsafeAtomicAdd(p, v);   // lowers to global_atomic_add_f32
}

// A-fragment from an LDS tile (row-major, stride ld): 16 rows starting m0, K=k0..k0+3
__device__ __forceinline__ v2f load_a_lds(const float* s, int ld, int m0, int k0,
                                          int mrow, int koff) {
  const float* p = s + (m0 + mrow) * ld + k0 + koff;
  v2f a; a.x = p[0]; a.y = p[1];
  return a;
}

// B-fragment = W^T tile. W is torch-convention [out,in] row-major with stride ldw.
// B[k][n] = W[n][k].  cols n0..n0+15, K=k0..k0+3.
__device__ __forceinline__ v2f load_b_wT(const float* Wp, int ldw, int n0, int k0,
                                         int ncol, int koff) {
  const float* p = Wp + (size_t)(n0 + ncol) * ldw + k0 + koff;
  v2f b; b.x = p[0]; b.y = p[1];
  return b;
}

// One 64x16 output block of a [64 x 128] @ W^T[128 x 16] GEMM.
// Independent A temporaries per k-step so 4 ds_loads can be in flight per wmma group.
__device__ __forceinline__ void gemm_k128(const float* __restrict__ sIn,
                                          const float* __restrict__ Wp,
                                          v8f acc[4], int n0, int mrow, int koff,
                                          int ncol16) {
  #pragma unroll 4
  for (int k0 = 0; k0 < HDIM; k0 += 4) {
    v2f b  = load_b_wT(Wp, HDIM, n0, k0, ncol16, koff);
    v2f a0 = load_a_lds(sIn, SROW,  0, k0, mrow, koff);
    v2f a1 = load_a_lds(sIn, SROW, 16, k0, mrow, koff);
    v2f a2 = load_a_lds(sIn, SROW, 32, k0, mrow, koff);
    v2f a3 = load_a_lds(sIn, SROW, 48, k0, mrow, koff);
    acc[0] = wmma_f32x4(a0, b, acc[0]);
    acc[1] = wmma_f32x4(a1, b, acc[1]);
    acc[2] = wmma_f32x4(a2, b, acc[2]);
    acc[3] = wmma_f32x4(a3, b, acc[3]);
  }
}

// ---------------------------------------------------------------------------
// Kernel 1: count unmasked walk steps per node / edge  (for scatter-mean)
// ---------------------------------------------------------------------------
__global__ void count_kernel(const int* __restrict__ nidx,
                             const int* __restrict__ eidx,
                             const unsigned char* __restrict__ nmask,
                             const unsigned char* __restrict__ emask,
                             float* __restrict__ ncnt,
                             float* __restrict__ ecnt, int total) {
  int i = blockIdx.x * blockDim.x + threadIdx.x;
  if (i >= total) return;
  if (!nmask[i]) atomAddF(&ncnt[nidx[i]], 1.0f);
  if (!emask[i]) atomAddF(&ecnt[eidx[i]], 1.0f);
}

// ---------------------------------------------------------------------------
// Kernel 2: fully fused per-walk pipeline (block = 1 walk, 256 thr = 8 waves).
//   gather(b128) -> (edge_proj + pe_proj) WMMA -> LayerNorm ->
//   depthwise conv + BN + ReLU -> pointwise WMMA + ReLU -> atomic scatter.
// ---------------------------------------------------------------------------
__global__ __launch_bounds__(256) void walk_kernel(
    const float* __restrict__ x, const float* __restrict__ edge_attr,
    const float* __restrict__ walk_pe,
    const int* __restrict__ wnidx, const int* __restrict__ weidx,
    const unsigned char* __restrict__ wnmask, const unsigned char* __restrict__ wemask,
    const float* __restrict__ Wep, const float* __restrict__ bep,
    const float* __restrict__ Wpe, const float* __restrict__ bpe,
    const float* __restrict__ ln_g, const float* __restrict__ ln_b,
    const float* __restrict__ dw_w, const float* __restrict__ dw_b,
    const float* __restrict__ bn1_g, const float* __restrict__ bn1_b,
    const float* __restrict__ bn1_m, const float* __restrict__ bn1_v,
    const float* __restrict__ Wpw, const float* __restrict__ bpw,
    float* __restrict__ node_sum, float* __restrict__ edge_sum) {
  extern __shared__ float smem[];
  float* sA = smem;                       // ROWS*SROW : x tile -> embedded walk
  float* sE = sA + ROWS * SROW;           // ROWS*SROW : edge tile -> conv output
  float* sP = sE + ROWS * SROW;           // ROWS*SPROW: PE tile (padded K=16)
  int*   sNidx = (int*)(sP + ROWS * SPROW);
  int*   sEidx = sNidx + ROWS;
  int*   sNm   = sEidx + ROWS;
  int*   sEm   = sNm + ROWS;

  const int tid  = threadIdx.x;
  const int w    = blockIdx.x;
  const int lane = tid & 31;
  const int wave = tid >> 5;
  const int mrow = lane & 15;             // A row within 16 / B col within 16
  const int koff = (lane >> 4) << 1;      // lanes 16-31 hold K=k0+2,k0+3
  const int mbase = (lane >> 4) << 3;     // C rows: lanes 16-31 hold M=8..15
  const int ncol16 = lane & 15;

  // ---- phase 0: walk-step indices + masks ----
  if (tid < ROWS) {
    if (tid < LWALK) {
      size_t o = (size_t)w * LWALK + tid;
      sNidx[tid] = wnidx[o];  sEidx[tid] = weidx[o];
      sNm[tid]   = wnmask[o]; sEm[tid]   = wemask[o];
    } else {
      sNidx[tid] = 0; sEidx[tid] = 0; sNm[tid] = 1; sEm[tid] = 1;
    }
  }
  __syncthreads();

  // ---- phase 1: gather masked rows into LDS (128-bit loads/stores) ----
  {
    const float4 z4 = make_float4(0.f, 0.f, 0.f, 0.f);
    for (int i = tid; i < ROWS * (HDIM / 4); i += 256) {
      int r = i >> 5, c4 = (i & 31) << 2;           // row, float offset
      float4 xv = z4, ev = z4;
      if (r < LWALK) {
        if (!sNm[r]) xv = *(const float4*)(x + (size_t)sNidx[r] * HDIM + c4);
        if (!sEm[r]) ev = *(const float4*)(edge_attr + (size_t)sEidx[r] * HDIM + c4);
      }
      *(float4*)(sA + r * SROW + c4) = xv;
      *(float4*)(sE + r * SROW + c4) = ev;
    }
    for (int i = tid; i < ROWS * 16; i += 256) {
      int r = i >> 4, c = i & 15;
      float pv = 0.f;
      if (r < LWALK && c < PEDIM)
        pv = walk_pe[((size_t)w * LWALK + r) * PEDIM + c];
      sP[r * SPROW + c] = pv;
    }
  }
  __syncthreads();

  // ---- phase 2: walk = x + edge @ Wep^T + pe @ Wpe^T + biases (WMMA) ----
  {
    const int n0 = wave * 16;
    v8f acc[4];
    #pragma unroll
    for (int mt = 0; mt < 4; ++mt) acc[mt] = (v8f)(0.0f);

    gemm_k128(sE, Wep, acc, n0, mrow, koff, ncol16);

    // PE projection, K padded 15 -> 16 (guard the pad column in B)
    #pragma unroll
    for (int k0 = 0; k0 < 16; k0 += 4) {
      int k = k0 + koff;
      v2f b;
      b.x = (k     < PEDIM) ? Wpe[(size_t)(n0 + ncol16) * PEDIM + k]     : 0.f;
      b.y = (k + 1 < PEDIM) ? Wpe[(size_t)(n0 + ncol16) * PEDIM + k + 1] : 0.f;
      v2f a0 = load_a_lds(sP, SPROW,  0, k0, mrow, koff);
      v2f a1 = load_a_lds(sP, SPROW, 16, k0, mrow, koff);
      v2f a2 = load_a_lds(sP, SPROW, 32, k0, mrow, koff);
      v2f a3 = load_a_lds(sP, SPROW, 48, k0, mrow, koff);
      acc[0] = wmma_f32x4(a0, b, acc[0]);
      acc[1] = wmma_f32x4(a1, b, acc[1]);
      acc[2] = wmma_f32x4(a2, b, acc[2]);
      acc[3] = wmma_f32x4(a3, b, acc[3]);
    }
    const int nc = n0 + ncol16;
    const float bias = bep[nc] + bpe[nc];
    #pragma unroll
    for (int mt = 0; mt < 4; ++mt)
      #pragma unroll
      for (int v = 0; v < 8; ++v) {
        int m = mt * 16 + mbase + v;
        sA[m * SROW + nc] += acc[mt][v] + bias;   // += residual x already in sA
      }
  }
  __syncthreads();

  // ---- phase 3: LayerNorm over H (one wave per row) ----
  for (int r = wave; r < LWALK; r += 8) {
    float vals[4], s = 0.f, sq = 0.f;
    #pragma unroll
    for (int j = 0; j < 4; ++j) {
      float u = sA[r * SROW + lane + j * 32];
      vals[j] = u; s += u; sq += u * u;
    }
    #pragma unroll
    for (int off = 16; off > 0; off >>= 1) {
      s  += __shfl_xor(s,  off, 32);
      sq += __shfl_xor(sq, off, 32);
    }
    float mu  = s * (1.0f / HDIM);
    float var = sq * (1.0f / HDIM) - mu * mu;
    float rs  = __frsqrt_rn(var + EPSV);
    #pragma unroll
    for (int j = 0; j < 4; ++j) {
      int c = lane + j * 32;
      sA[r * SROW + c] = (vals[j] - mu) * rs * ln_g[c] + ln_b[c];
    }
  }
  __syncthreads();

  // ---- phase 4: depthwise conv(K=9, same pad) + BN(eval) + ReLU -> sE ----
  {
    int h = tid & 127, half = tid >> 7;
    float wk[KCONV];
    #pragma unroll
    for (int j = 0; j < KCONV; ++j) wk[j] = dw_w[h * KCONV + j];
    float scale = bn1_g[h] * __frsqrt_rn(bn1_v[h] + EPSV);
    float shift = bn1_b[h] + (dw_b[h] - bn1_m[h]) * scale;
    for (int l = half; l < LWALK; l += 2) {
      float s = 0.f;
      #pragma unroll
      for (int j = 0; j < KCONV; ++j) {
        int ll = l + j - (KCONV / 2);
        if (ll >= 0 && ll < LWALK) s += wk[j] * sA[ll * SROW + h];
      }
      sE[l * SROW + h] = fmaxf(s * scale + shift, 0.f);
    }
    for (int l = LWALK + half; l < ROWS; l += 2) sE[l * SROW + h] = 0.f;
  }
  __syncthreads();

  // ---- phase 5: pointwise GEMM + ReLU + atomic scatter-add ----
  {
    const int n0 = wave * 16;
    v8f acc[4];
    #pragma unroll
    for (int mt = 0; mt < 4; ++mt) acc[mt] = (v8f)(0.0f);
    gemm_k128(sE, Wpw, acc, n0, mrow, koff, ncol16);

    const int nc = n0 + ncol16;
    const float pb = bpw[nc];
    #pragma unroll
    for (int mt = 0; mt < 4; ++mt)
      #pragma unroll
      for (int v = 0; v < 8; ++v) {
        int m = mt * 16 + mbase + v;
        if (m < LWALK) {
          float val = fmaxf(acc[mt][v] + pb, 0.f);
          if (!sNm[m]) atomAddF(&node_sum[(size_t)sNidx[m] * HDIM + nc], val);
          if (!sEm[m]) atomAddF(&edge_sum[(size_t)sEidx[m] * HDIM + nc], val);
        }
      }
  }
}

// ---------------------------------------------------------------------------
// Kernel 3: out = base + MLP(sum/max(cnt,1)); MLP = W1 -> BN -> ReLU -> W2
//   64 rows per block, two WMMA GEMMs through LDS.
// ---------------------------------------------------------------------------
__global__ __launch_bounds__(256) void agg_mlp_kernel(
    const float* __restrict__ sum, const float* __restrict__ cnt,
    const float* __restrict__ base,
    const float* __restrict__ W1, const float* __restrict__ b1,
    const float* __restrict__ g,  const float* __restrict__ bb,
    const float* __restrict__ mn, const float* __restrict__ vv,
    const float* __restrict__ W2, const float* __restrict__ b2,
    float* __restrict__ out, int rows) {
  extern __shared__ float smem[];
  float* sIn  = smem;                 // ROWS*SROW
  float* sMid = sIn + ROWS * SROW;    // ROWS*SROW
  float* sRcp = sMid + ROWS * SROW;   // ROWS per-row 1/max(cnt,1)

  const int tid  = threadIdx.x;
  const int lane = tid & 31;
  const int wave = tid >> 5;
  const int mrow = lane & 15;
  const int koff = (lane >> 4) << 1;
  const int mbase = (lane >> 4) << 3;
  const int ncol16 = lane & 15;
  const int r0 = blockIdx.x * ROWS;

  if (tid < ROWS) {
    int gr = r0 + tid;
    sRcp[tid] = (gr < rows) ? (1.0f / fmaxf(cnt[gr], 1.0f)) : 0.0f;
  }
  __syncthreads();

  {
    const float4 z4 = make_float4(0.f, 0.f, 0.f, 0.f);
    for (int i = tid; i < ROWS * (HDIM / 4); i += 256) {
      int r = i >> 5, c4 = (i & 31) << 2;
      int gr = r0 + r;
      float4 v = z4;
      if (gr < rows) {
        v = *(const float4*)(sum + (size_t)gr * HDIM + c4);
        float rc = sRcp[r];
        v.x *= rc; v.y *= rc; v.z *= rc; v.w *= rc;
      }
      *(float4*)(sIn + r * SROW + c4) = v;
    }
  }
  __syncthreads();

  // GEMM1 + BN + ReLU -> sMid
  {
    const int n0 = wave * 16;
    v8f acc[4];
    #pragma unroll
    for (int mt = 0; mt < 4; ++mt) acc[mt] = (v8f)(0.0f);
    gemm_k128(sIn, W1, acc, n0, mrow, koff, ncol16);

    const int nc = n0 + ncol16;
    float scale = g[nc] * __frsqrt_rn(vv[nc] + EPSV);
    float shift = bb[nc] + (b1[nc] - mn[nc]) * scale;
    #pragma unroll
    for (int mt = 0; mt < 4; ++mt)
      #pragma unroll
      for (int v = 0; v < 8; ++v) {
        int m = mt * 16 + mbase + v;
        sMid[m * SROW + nc] = fmaxf(acc[mt][v] * scale + shift, 0.f);
      }
  }
  __syncthreads();

  // GEMM2 + bias + residual -> out
  {
    const int n0 = wave * 16;
    v8f acc[4];
    #pragma unroll
    for (int mt = 0; mt < 4; ++mt) acc[mt] = (v8f)(0.0f);
    gemm_k128(sMid, W2, acc, n0, mrow, koff, ncol16);

    const int nc = n0 + ncol16;
    const float bias = b2[nc];
    #pragma unroll
    for (int mt = 0; mt < 4; ++mt)
      #pragma unroll
      for (int v = 0; v < 8; ++v) {
        int m = mt * 16 + mbase + v;
        int gr = r0 + m;
        if (gr < rows)
          out[(size_t)gr * HDIM + nc] =
              acc[mt][v] + bias + base[(size_t)gr * HDIM + nc];
      }
  }
}

// ---------------------------------------------------------------------------
extern "C" void kernel_launch(void* const* d_in, const int* in_sizes, int n_in,
                              void* d_out, int out_size, void* d_ws, size_t ws_size,
                              hipStream_t stream) {
  // inputs in setup_inputs() order
  const float* x          = (const float*)d_in[0];
  const float* edge_attr  = (const float*)d_in[1];
  const float* walk_pe    = (const float*)d_in[2];
  const int*   wnidx      = (const int*)d_in[3];
  const int*   weidx      = (const int*)d_in[4];
  const unsigned char* wnmask = (const unsigned char*)d_in[5];  // jax bool -> u8
  const unsigned char* wemask = (const unsigned char*)d_in[6];
  const float* Wep  = (const float*)d_in[7];
  const float* bep  = (const float*)d_in[8];
  const float* Wpe  = (const float*)d_in[9];
  const float* bpe  = (const float*)d_in[10];
  const float* ln_g = (const float*)d_in[11];
  const float* ln_b = (const float*)d_in[12];
  const float* dw_w = (const float*)d_in[13];
  const float* dw_b = (const float*)d_in[14];
  const float* bn1_g = (const float*)d_in[15];
  const float* bn1_b = (const float*)d_in[16];
  const float* bn1_m = (const float*)d_in[17];
  const float* bn1_v = (const float*)d_in[18];
  const float* Wpw  = (const float*)d_in[19];
  const float* bpw  = (const float*)d_in[20];
  const float* nW1  = (const float*)d_in[21];
  const float* nb1  = (const float*)d_in[22];
  const float* nbg  = (const float*)d_in[23];
  const float* nbb  = (const float*)d_in[24];
  const float* nbm  = (const float*)d_in[25];
  const float* nbv  = (const float*)d_in[26];
  const float* nW2  = (const float*)d_in[27];
  const float* nb2  = (const float*)d_in[28];
  const float* eW1  = (const float*)d_in[29];
  const float* eb1  = (const float*)d_in[30];
  const float* ebg  = (const float*)d_in[31];
  const float* ebb  = (const float*)d_in[32];
  const float* ebm  = (const float*)d_in[33];
  const float* ebv  = (const float*)d_in[34];
  const float* eW2  = (const float*)d_in[35];
  const float* eb2  = (const float*)d_in[36];

  float* out = (float*)d_out;

  // workspace layout (floats): node_sum | edge_sum | node_cnt | edge_cnt
  float* ws = (float*)d_ws;
  float* node_sum = ws;
  float* edge_sum = node_sum + (size_t)NNODE * HDIM;
  float* node_cnt = edge_sum + (size_t)NEDGE * HDIM;
  float* edge_cnt = node_cnt + NNODE;
  size_t zero_bytes =
      ((size_t)NNODE * HDIM + (size_t)NEDGE * HDIM + NNODE + NEDGE) * sizeof(float);
  hipMemsetAsync(d_ws, 0, zero_bytes, stream);

  // 1) counts for scatter-mean
  {
    int total = NWALK * LWALK;
    count_kernel<<<(total + 255) / 256, 256, 0, stream>>>(
        wnidx, weidx, wnmask, wemask, node_cnt, edge_cnt, total);
  }

  // 2) fused per-walk pipeline
  {
    size_t lds = (size_t)(2 * ROWS * SROW + ROWS * SPROW) * sizeof(float) +
                 (size_t)(4 * ROWS) * sizeof(int);
    walk_kernel<<<NWALK, 256, lds, stream>>>(
        x, edge_attr, walk_pe, wnidx, weidx, wnmask, wemask,
        Wep, bep, Wpe, bpe, ln_g, ln_b, dw_w, dw_b,
        bn1_g, bn1_b, bn1_m, bn1_v, Wpw, bpw, node_sum, edge_sum);
  }

  // 3) node & edge residual MLPs -> d_out (x_out then e_out, concatenated)
  {
    size_t lds = (size_t)(2 * ROWS * SROW + ROWS) * sizeof(float);
    agg_mlp_kernel<<<(NNODE + ROWS - 1) / ROWS, 256, lds, stream>>>(
        node_sum, node_cnt, x, nW1, nb1, nbg, nbb, nbm, nbv, nW2, nb2,
        out, NNODE);
    agg_mlp_kernel<<<(NEDGE + ROWS - 1) / ROWS, 256, lds, stream>>>(
        edge_sum, edge_cnt, edge_attr, eW1, eb1, ebg, ebb, ebm, ebv, eW2, eb2,
        out + (size_t)NNODE * HDIM, NEDGE);
  }
}